// BaselineRemove_46437186404848
// MI455X (gfx1250) — compile-verified
//
#include <hip/hip_runtime.h>
#include <stdint.h>

// Problem constants (match reference: FS=500 -> K1=101, K2=301; T=4000)
#define TT   4000
#define R1   50      // stage-1 radius (k=101)
#define K1W  101
#define R2   150     // stage-2 radius (k=301)
#define K2W  301
#define HALO (R1 + R2)          // 200: x halo needed per output tile
#define TS   500                // outputs per block (TT % TS == 0)
#define TPR  (TT / TS)          // 8 tiles per row (compile-time: no SALU divide)
#define XW   (TS + 2 * HALO)    // 900: x samples staged per block
#define YW   (TS + 2 * R2)      // 800: stage-1 medians needed per block
#define BLK  256                // 8 wave32s

// Monotone float -> uint32 key (order-preserving), and inverse.
__device__ __forceinline__ uint32_t f2key(uint32_t u) {
  return (u & 0x80000000u) ? ~u : (u | 0x80000000u);
}
__device__ __forceinline__ float key2f(uint32_t k) {
  uint32_t u = (k & 0x80000000u) ? (k & 0x7FFFFFFFu) : ~k;
  return __uint_as_float(u);
}

// MSB radix selection of the `rank`-th smallest (0-indexed) of k keys in LDS,
// 2 bits per pass (16 passes). An element is still in play with pair value p
// iff (key ^ prefix) >> bit == p (prefix has zeros below the decided bits),
// so the inner loop is 1 ds_load + xor + lshr + 3x(cmp,add) per element.
__device__ __forceinline__ uint32_t radix_select(const uint32_t* __restrict__ w,
                                                 int k, int rank) {
  uint32_t prefix = 0u;
  int need = rank;
#pragma unroll 1
  for (int bit = 30; bit >= 0; bit -= 2) {
    int c0 = 0, c1 = 0, c2 = 0;
#pragma unroll 4
    for (int j = 0; j < k; ++j) {
      const uint32_t u = (w[j] ^ prefix) >> bit;
      c0 += (int)(u == 0u);
      c1 += (int)(u == 1u);
      c2 += (int)(u == 2u);
    }
    const int c01 = c0 + c1, c012 = c01 + c2;
    uint32_t p;
    if      (need >= c012) { p = 3u; need -= c012; }
    else if (need >= c01)  { p = 2u; need -= c01;  }
    else if (need >= c0)   { p = 1u; need -= c0;   }
    else                   { p = 0u; }
    prefix |= p << bit;
  }
  return prefix;
}

__global__ __launch_bounds__(BLK) void BaselineRemove_kernel(
    const float* __restrict__ sig, float* __restrict__ out) {
  __shared__ uint32_t kxs[XW];  // staged x (raw bits, then keys in-place)
  __shared__ uint32_t kys[YW];  // stage-1 median keys

  const int tile = blockIdx.x % TPR;          // compile-time modulus -> and/shift
  const int row  = blockIdx.x / TPR;
  const int t0   = tile * TS;                 // first output sample of this tile
  const int tid  = threadIdx.x;
  const float* src = sig + (size_t)row * TT;

  // ---- Phase A: async global -> LDS stage of x tile + halo (edge-clamped) --
  for (int i = tid; i < XW; i += BLK) {
    int gi = t0 - HALO + i;
    gi = gi < 0 ? 0 : (gi > TT - 1 ? TT - 1 : gi);   // edge replication
    uint64_t ga = (uint64_t)(uintptr_t)(src + gi);
    uint32_t la = (uint32_t)(uintptr_t)&kxs[i];      // low 32 bits = LDS offset
    asm volatile("global_load_async_to_lds_b32 %0, %1, off"
                 :: "v"(la), "v"(ga) : "memory");
  }
  asm volatile("s_wait_asynccnt 0" ::: "memory");    // ASYNCcnt drain (per wave)
  __syncthreads();

  // ---- Phase B: convert to monotone keys in-place ------------------------
  for (int i = tid; i < XW; i += BLK)
    kxs[i] = f2key(kxs[i]);
  __syncthreads();

  // ---- Phase C: stage-1 medians (k=101) for tile +/- R2 ------------------
  // kys[yi] = y[clamp(t0 - R2 + yi)] so stage-2 edge padding is exact.
  for (int yi = tid; yi < YW; yi += BLK) {
    int tg = t0 - R2 + yi;
    int tc = tg < 0 ? 0 : (tg > TT - 1 ? TT - 1 : tg);
    // kxs[j] holds x[clamp(t0 - HALO + j)]; window start x[tc - R1]:
    const int js = tc - t0 + HALO - R1;              // in [0, XW - K1W]
    kys[yi] = radix_select(&kxs[js], K1W, R1);
  }
  __syncthreads();

  // ---- Phase D: stage-2 medians (k=301) + subtraction --------------------
  for (int i = tid; i < TS; i += BLK) {
    const uint32_t zk = radix_select(&kys[i], K2W, R2);
    const float x = key2f(kxs[i + HALO]);            // x[t0 + i]
    out[(size_t)row * TT + t0 + i] = x - key2f(zk);
  }
}

extern "C" void kernel_launch(void* const* d_in, const int* in_sizes, int n_in,
                              void* d_out, int out_size, void* d_ws, size_t ws_size,
                              hipStream_t stream) {
  const float* sig = (const float*)d_in[0];
  float* out = (float*)d_out;
  const int rows = in_sizes[0] / TT;        // 8*12 = 96
  dim3 grid(rows * TPR), block(BLK);
  hipLaunchKernelGGL(BaselineRemove_kernel, grid, block, 0, stream, sig, out);
}